// KF_49830210568555
// MI455X (gfx1250) — compile-verified
//
#include <hip/hip_runtime.h>

// Kalman filter for MI455X (gfx1250, wave32, WMMA).
//   Kernel 1 (kf_riccati): 1 block x 1024 thr (one WGP), sequential Riccati
//     recursion; all matmuls via V_WMMA_F32_16X16X4_F32 with BOTH operands in
//     LDS (F/H staged once); Newton-Schulz inverse of S_obs (warm-started).
//     Emits Ps, Sos to d_out; K_t stream -> d_ws (16MB, L2-resident).
//   Kernel 2 (kf_batch): 64 blocks x 256 thr, 16 batch rows/block over all T.
//     Software-pipelined: u/y(t+1) reg-prefetch, K_t coalesced reg->LDS stage
//     overlapped with predict WMMAs, global_prefetch for K_{t+1}.

typedef __attribute__((ext_vector_type(2))) float v2f;
typedef __attribute__((ext_vector_type(8))) float v8f;

#define S_DIM 128
#define I_DIM 64
#define O_DIM 64
#define BATCH_N 1024
#define T_N 512

static __device__ __forceinline__ v8f v8f_zero() { v8f z = {}; return z; }

static __device__ __forceinline__ v8f wmma_f32(v2f a, v2f b, v8f c) {
  // D = A(16x4) * B(4x16) + C(16x16), fp32 throughout.
  return __builtin_amdgcn_wmma_f32_16x16x4_f32(
      /*neg_a=*/false, a, /*neg_b=*/false, b,
      /*c_mod=*/(short)0, c, /*reuse_a=*/false, /*reuse_b=*/false);
}

// 16x16 output tile at (m0,n0), accumulating over K (multiple of 4).
// AT=false: A row-major [M x K]  -> A[m][k] = A[m*lda+k]
// AT=true : A given as [K x M]   -> A[m][k] = A[k*lda+m]   (transposed read)
// BT=false: B row-major [K x N]  -> B[k][n] = B[k*ldb+n]
// BT=true : B = W^T, W row-major [N x K] -> B[k][n] = W[n*ldb+k]
template <bool AT, bool BT>
static __device__ __forceinline__ v8f mma_tile(const float* __restrict__ A, int lda,
                                               const float* __restrict__ B, int ldb,
                                               int m0, int n0, int K, v8f acc) {
  const int l16  = threadIdx.x & 15;
  const int half = (threadIdx.x >> 4) & 1;   // which K-pair this lane holds
  const int am = m0 + l16;
  const int bn = n0 + l16;
#pragma unroll 4
  for (int k = 0; k < K; k += 4) {
    const int ka = k + 2 * half;             // lanes 0-15: K=k,k+1; 16-31: k+2,k+3
    v2f a, b;
    if (AT) { a[0] = A[ka * lda + am];       a[1] = A[(ka + 1) * lda + am]; }
    else    { a[0] = A[am * lda + ka];       a[1] = A[am * lda + ka + 1];  }
    if (BT) { b[0] = B[bn * ldb + ka];       b[1] = B[bn * ldb + ka + 1];  }
    else    { b[0] = B[ka * ldb + bn];       b[1] = B[(ka + 1) * ldb + bn];}
    acc = wmma_f32(a, b, acc);
  }
  return acc;
}

// ---------------------------------------------------------------------------
// Kernel 1: sequential Riccati recursion (one workgroup = one WGP).
// LDS: Pm(64K) Tm(64K) Fs(64K) Hs(32K) HP/Km(32K) Sm(16K) Xi(16K) T2(16K)
//      = 304 KB  (<= 320 KB per-workgroup LDS on CDNA5).
// ---------------------------------------------------------------------------
__global__ void __launch_bounds__(1024)
kf_riccati(const float* __restrict__ F, const float* __restrict__ H,
           const float* __restrict__ SW, const float* __restrict__ SV,
           float* __restrict__ Ps_out, float* __restrict__ Sos_out,
           float* __restrict__ Kg) {
  extern __shared__ float smem[];
  float* Pm = smem;            // 128x128 current/pred covariance
  float* Tm = Pm + 16384;      // 128x128 temp (F@P, then K@H)
  float* Fs = Tm + 16384;      // 128x128 staged F
  float* Hs = Fs + 16384;      // 64x128  staged H
  float* HP = Hs + 8192;       // 64x128  H@P'   (aliased by Km after phase 6)
  float* Sm = HP + 8192;       // 64x64   innovation covariance
  float* Xi = Sm + 4096;       // 64x64   running inverse (warm-started)
  float* T2 = Xi + 4096;       // 64x64   Newton temp
  float* Km = HP;              // 128x64  Kalman gain (reuses HP storage)

  const int tid  = threadIdx.x;
  const int w    = tid >> 5;              // wave id 0..31
  const int l16  = tid & 15;
  const int half = (tid >> 4) & 1;

  for (int i = tid; i < 16384; i += 1024) { Pm[i] = 0.0f; Fs[i] = F[i]; }
  for (int i = tid; i <  8192; i += 1024) Hs[i] = H[i];
  __syncthreads();

  for (int t = 0; t < T_N; ++t) {
    // ---- Tm = F @ P  (64 tiles, 2 per wave) ----
#pragma unroll
    for (int i = 0; i < 2; ++i) {
      const int tt = w * 2 + i, m0 = (tt >> 3) * 16, n0 = (tt & 7) * 16;
      v8f c = v8f_zero();
      c = mma_tile<false, false>(Fs, 128, Pm, 128, m0, n0, 128, c);
#pragma unroll
      for (int r = 0; r < 8; ++r) Tm[(m0 + half * 8 + r) * 128 + n0 + l16] = c[r];
    }
    __syncthreads();
    // ---- P' = Tm @ F^T + S_W ----
#pragma unroll
    for (int i = 0; i < 2; ++i) {
      const int tt = w * 2 + i, m0 = (tt >> 3) * 16, n0 = (tt & 7) * 16;
      v8f c = v8f_zero();
      c = mma_tile<false, true>(Tm, 128, Fs, 128, m0, n0, 128, c);
#pragma unroll
      for (int r = 0; r < 8; ++r) {
        const int idx = (m0 + half * 8 + r) * 128 + n0 + l16;
        Pm[idx] = c[r] + SW[idx];
      }
    }
    __syncthreads();
    // ---- HP = H @ P'  (32 tiles, 1 per wave) ----
    {
      const int m0 = (w >> 3) * 16, n0 = (w & 7) * 16;
      v8f c = v8f_zero();
      c = mma_tile<false, false>(Hs, 128, Pm, 128, m0, n0, 128, c);
#pragma unroll
      for (int r = 0; r < 8; ++r) HP[(m0 + half * 8 + r) * 128 + n0 + l16] = c[r];
    }
    __syncthreads();
    // ---- S = HP @ H^T + S_V ; emit Sos[t] ----
    if (w < 16) {
      const int m0 = (w >> 2) * 16, n0 = (w & 3) * 16;
      v8f c = v8f_zero();
      c = mma_tile<false, true>(HP, 128, Hs, 128, m0, n0, 128, c);
#pragma unroll
      for (int r = 0; r < 8; ++r) {
        const int idx = (m0 + half * 8 + r) * 64 + n0 + l16;
        const float v = c[r] + SV[idx];
        Sm[idx] = v;
        Sos_out[(size_t)t * 4096 + idx] = v;
      }
    }
    __syncthreads();
    // ---- Xi ~= inv(S): Newton-Schulz  X <- X(2I - S X), warm-started ----
    int nit = 6;
    if (t == 0) {
      nit = 30;                                    // cold start: X0 = S / ||S||_1^2
      if (tid < 64) {
        float rs = 0.0f;
        for (int cc = 0; cc < 64; ++cc) rs += fabsf(Sm[tid * 64 + cc]);
        T2[tid] = rs;
      }
      __syncthreads();
      if (tid == 0) {
        float mx = 0.0f;
        for (int r = 0; r < 64; ++r) mx = fmaxf(mx, T2[r]);
        T2[64] = 1.0f / (mx * mx);
      }
      __syncthreads();
      const float sc = T2[64];
      for (int i = tid; i < 4096; i += 1024) Xi[i] = Sm[i] * sc;
      __syncthreads();
    }
    for (int it = 0; it < nit; ++it) {
      if (w < 16) {                                // T2 = S @ Xi
        const int m0 = (w >> 2) * 16, n0 = (w & 3) * 16;
        v8f c = v8f_zero();
        c = mma_tile<false, false>(Sm, 64, Xi, 64, m0, n0, 64, c);
#pragma unroll
        for (int r = 0; r < 8; ++r) T2[(m0 + half * 8 + r) * 64 + n0 + l16] = c[r];
      }
      __syncthreads();
      v8f c = v8f_zero();
      int m0 = 0, n0 = 0;
      if (w < 16) {                                // c = Xi @ T2 (into regs)
        m0 = (w >> 2) * 16; n0 = (w & 3) * 16;
        c = mma_tile<false, false>(Xi, 64, T2, 64, m0, n0, 64, c);
      }
      __syncthreads();
      if (w < 16) {                                // Xi = 2*Xi - c
#pragma unroll
        for (int r = 0; r < 8; ++r) {
          const int idx = (m0 + half * 8 + r) * 64 + n0 + l16;
          Xi[idx] = 2.0f * Xi[idx] - c[r];
        }
      }
      __syncthreads();
    }
    // ---- K = (HP)^T @ Xi  [P' symmetric => P'H^T = (HP)^T] ----
    // Compute into regs + write global, barrier (HP reads done), then park the
    // gain in LDS by aliasing the now-dead HP buffer.
    {
      const int m0 = (w >> 2) * 16, n0 = (w & 3) * 16;   // 8x4 tiles, 1 per wave
      v8f ck = v8f_zero();
      ck = mma_tile<true, false>(HP, 128, Xi, 64, m0, n0, 64, ck);
      float* Kt = Kg + (size_t)t * (S_DIM * O_DIM);
#pragma unroll
      for (int r = 0; r < 8; ++r)
        Kt[(m0 + half * 8 + r) * 64 + n0 + l16] = ck[r];
      __syncthreads();                              // all reads of HP complete
#pragma unroll
      for (int r = 0; r < 8; ++r)
        Km[(m0 + half * 8 + r) * 64 + n0 + l16] = ck[r];
    }
    __syncthreads();
    // ---- Tm = K @ H ----
#pragma unroll
    for (int i = 0; i < 2; ++i) {
      const int tt = w * 2 + i, m0 = (tt >> 3) * 16, n0 = (tt & 7) * 16;
      v8f c = v8f_zero();
      c = mma_tile<false, false>(Km, 64, Hs, 128, m0, n0, 64, c);
#pragma unroll
      for (int r = 0; r < 8; ++r) Tm[(m0 + half * 8 + r) * 128 + n0 + l16] = c[r];
    }
    __syncthreads();
    // ---- P = P' - Tm @ P' ; emit Ps[t] ----
    v8f acc0 = v8f_zero(), acc1 = v8f_zero();
    {
      int tt = w * 2 + 0, m0 = (tt >> 3) * 16, n0 = (tt & 7) * 16;
      acc0 = mma_tile<false, false>(Tm, 128, Pm, 128, m0, n0, 128, acc0);
      tt = w * 2 + 1; m0 = (tt >> 3) * 16; n0 = (tt & 7) * 16;
      acc1 = mma_tile<false, false>(Tm, 128, Pm, 128, m0, n0, 128, acc1);
    }
    __syncthreads();
#pragma unroll
    for (int i = 0; i < 2; ++i) {
      const int tt = w * 2 + i, m0 = (tt >> 3) * 16, n0 = (tt & 7) * 16;
#pragma unroll
      for (int r = 0; r < 8; ++r) {
        const int idx = (m0 + half * 8 + r) * 128 + n0 + l16;
        const float v = Pm[idx] - (i == 0 ? acc0[r] : acc1[r]);
        Pm[idx] = v;
        Ps_out[(size_t)t * 16384 + idx] = v;
      }
    }
    __syncthreads();
  }
}

// ---------------------------------------------------------------------------
// Kernel 2: batch-parallel state recursion. 64 blocks x 256 thr (8 waves),
// 16 batch rows per block across all T steps. LDS ~184KB.
// Software pipeline: u/y(t+1) prefetched into regs during step t; K_t loaded
// coalesced into regs at step top (overlaps predict WMMAs), parked in LDS,
// consumed from LDS in the correction phase; K_{t+1} warmed via prefetch.
// ---------------------------------------------------------------------------
__global__ void __launch_bounds__(256)
kf_batch(const float* __restrict__ x0, const float* __restrict__ U,
         const float* __restrict__ Y,  const float* __restrict__ F,
         const float* __restrict__ Bm, const float* __restrict__ H,
         const float* __restrict__ Kg,
         float* __restrict__ Xout, float* __restrict__ Yhout) {
  extern __shared__ float smem[];
  float* Fs  = smem;            // 128x128
  float* Bs  = Fs + 16384;      // 128x64
  float* Hs  = Bs + 8192;       // 64x128
  float* Ks  = Hs + 8192;       // 128x64 staged K_t
  float* xs  = Ks + 8192;       // 16x128 current state
  float* xn  = xs + 2048;       // 16x128 predicted state
  float* us  = xn + 2048;       // 16x64  inputs
  float* res = us + 1024;       // 16x64  y, then residual

  const int tid  = threadIdx.x;
  const int w    = tid >> 5;              // wave 0..7
  const int l16  = tid & 15;
  const int half = (tid >> 4) & 1;
  const int r0   = blockIdx.x * 16;       // first batch row of this block

  for (int i = tid; i < 16384; i += 256) Fs[i] = F[i];
  for (int i = tid; i <  8192; i += 256) Bs[i] = Bm[i];
  for (int i = tid; i <  8192; i += 256) Hs[i] = H[i];
  for (int i = tid; i <  2048; i += 256) xs[i] = x0[(size_t)r0 * 128 + i];

  // Register prefetch of u_0 / y_0 (4 elements each per thread).
  float uf[4], yf[4];
#pragma unroll
  for (int i = 0; i < 4; ++i) {
    const int idx = i * 256 + tid, rr = idx >> 6, cc = idx & 63;
    uf[i] = U[((size_t)(r0 + rr) * T_N) * 64 + cc];
    yf[i] = Y[((size_t)(r0 + rr) * T_N) * 64 + cc];
  }
  __syncthreads();

  for (int t = 0; t < T_N; ++t) {
    // commit prefetched u_t / y_t to LDS
#pragma unroll
    for (int i = 0; i < 4; ++i) {
      us[i * 256 + tid]  = uf[i];
      res[i * 256 + tid] = yf[i];
    }
    // issue coalesced K_t loads into regs (latency overlaps predict WMMAs)
    const float* Kt = Kg + (size_t)t * (S_DIM * O_DIM);
    float kreg[32];
#pragma unroll
    for (int i = 0; i < 32; ++i) kreg[i] = Kt[i * 256 + tid];
    if (t + 1 < T_N)                       // warm L2/L0 for next gain matrix
      __builtin_prefetch(Kg + (size_t)(t + 1) * (S_DIM * O_DIM) + tid * 32, 0, 1);
    __syncthreads();
    // prefetch u/y for t+1 into regs
    if (t + 1 < T_N) {
#pragma unroll
      for (int i = 0; i < 4; ++i) {
        const int idx = i * 256 + tid, rr = idx >> 6, cc = idx & 63;
        uf[i] = U[((size_t)(r0 + rr) * T_N + (t + 1)) * 64 + cc];
        yf[i] = Y[((size_t)(r0 + rr) * T_N + (t + 1)) * 64 + cc];
      }
    }
    // ---- predict: xn = xs@F^T + us@B^T  (8 column tiles, 1 per wave) ----
    const int n0 = w * 16;
    v8f cx = v8f_zero();
    cx = mma_tile<false, true>(xs, 128, Fs, 128, 0, n0, 128, cx);
    cx = mma_tile<false, true>(us,  64, Bs,  64, 0, n0,  64, cx);
#pragma unroll
    for (int r = 0; r < 8; ++r) xn[(half * 8 + r) * 128 + n0 + l16] = cx[r];
    __syncthreads();
    // park K_t in LDS (loads have had the whole predict phase to land)
#pragma unroll
    for (int i = 0; i < 32; ++i) Ks[i * 256 + tid] = kreg[i];
    // ---- y_hat = xn@H^T (waves 0..3); residual; emit observation est ----
    if (w < 4) {
      const int o0 = w * 16;
      v8f cy = v8f_zero();
      cy = mma_tile<false, true>(xn, 128, Hs, 128, 0, o0, 128, cy);
#pragma unroll
      for (int r = 0; r < 8; ++r) {
        const int row = half * 8 + r, col = o0 + l16;
        const float yh = cy[r];
        Yhout[((size_t)(r0 + row) * T_N + t) * 64 + col] = yh;
        res[row * 64 + col] -= yh;                  // y - y_hat
      }
    }
    __syncthreads();                                // covers res and Ks
    // ---- correct: x = xn + res @ K_t^T ; emit state est ----
    cx = mma_tile<false, true>(res, 64, Ks, 64, 0, n0, 64, cx);
#pragma unroll
    for (int r = 0; r < 8; ++r) {
      const int row = half * 8 + r, col = n0 + l16;
      xs[row * 128 + col] = cx[r];
      Xout[((size_t)(r0 + row) * T_N + t) * 128 + col] = cx[r];
    }
    __syncthreads();
  }
}

// ---------------------------------------------------------------------------
extern "C" void kernel_launch(void* const* d_in, const int* in_sizes, int n_in,
                              void* d_out, int out_size, void* d_ws, size_t ws_size,
                              hipStream_t stream) {
  const float* x0 = (const float*)d_in[0];   // [1024,128]
  const float* U  = (const float*)d_in[1];   // [1024,512,64]
  const float* Y  = (const float*)d_in[2];   // [1024,512,64]
  const float* F  = (const float*)d_in[3];   // [128,128]
  const float* Bm = (const float*)d_in[4];   // [128,64]
  const float* H  = (const float*)d_in[5];   // [64,128]
  const float* SW = (const float*)d_in[6];   // [128,128]
  const float* SV = (const float*)d_in[7];   // [64,64]
  float* out = (float*)d_out;
  float* Kg  = (float*)d_ws;                 // needs 512*128*64*4 = 16 MB

  // d_out layout: state_est [B,T,S] | Ps [T,S,S] | obs_est [B,T,O] | Sos [T,O,O]
  const size_t PS_OFF  = (size_t)BATCH_N * T_N * S_DIM;   // 67,108,864
  const size_t OE_OFF  = PS_OFF + (size_t)T_N * S_DIM * S_DIM;
  const size_t SOS_OFF = OE_OFF + (size_t)BATCH_N * T_N * O_DIM;

  const size_t shA = (size_t)(16384 * 3 + 8192 * 2 + 4096 * 3) * 4;  // 304 KB
  const size_t shB = (size_t)(16384 + 8192 * 3 + 2048 * 2 + 1024 * 2) * 4;
  hipFuncSetAttribute((const void*)kf_riccati,
                      hipFuncAttributeMaxDynamicSharedMemorySize, (int)shA);
  hipFuncSetAttribute((const void*)kf_batch,
                      hipFuncAttributeMaxDynamicSharedMemorySize, (int)shB);

  kf_riccati<<<1, 1024, shA, stream>>>(F, H, SW, SV,
                                       out + PS_OFF, out + SOS_OFF, Kg);
  kf_batch<<<BATCH_N / 16, 256, shB, stream>>>(x0, U, Y, F, Bm, H, Kg,
                                               out, out + OE_OFF);
}